// net_34273839022738
// MI455X (gfx1250) — compile-verified
//
#include <hip/hip_runtime.h>
#include <math.h>

// ---------------------------------------------------------------------------
// CapsNet forward for MI455X (gfx1250, wave32).
// Dominant cost: streaming W (3.96 GB) once + u_hat (496 MB) 1W+2R ≈ 6 GB
// total HBM traffic -> ~260 us floor at 23.3 TB/s. PrimaryCaps conv is the
// compute-dense stage and runs on V_WMMA_F32_16X16X4_F32.
// ---------------------------------------------------------------------------

#define NNODES 774400            // 16*220*220
#define NP     48400             // 220*220
#define A1     49284             // 222*222
#define JST    ((size_t)NNODES * 16)

typedef __attribute__((ext_vector_type(2))) float v2f;
typedef __attribute__((ext_vector_type(8))) float v8f;

__device__ __forceinline__ void atomAddG(float* p, float v) {
  __hip_atomic_fetch_add(p, v, __ATOMIC_RELAXED, __HIP_MEMORY_SCOPE_AGENT);
}
__device__ __forceinline__ void atomAddL(float* p, float v) {
  __hip_atomic_fetch_add(p, v, __ATOMIC_RELAXED, __HIP_MEMORY_SCOPE_WORKGROUP);
}

// ---------------------------------------------------------------- zero s1..s3
__global__ void zero_kernel(float* p, int n) {
  int i = blockIdx.x * blockDim.x + threadIdx.x;
  if (i < n) p[i] = 0.0f;
}

// ---------------------------------------------------- Conv1 3->32 3x3 + ReLU
__global__ void conv1_relu(const float* __restrict__ img,
                           const float* __restrict__ w,
                           const float* __restrict__ b,
                           float* __restrict__ feats) {
  int idx = blockIdx.x * 256 + threadIdx.x;
  if (idx >= 32 * A1) return;
  int oc = idx / A1;
  int p  = idx - oc * A1;
  int y  = p / 222;
  int x  = p - y * 222;
  float acc = b[oc];
#pragma unroll
  for (int ic = 0; ic < 3; ++ic)
#pragma unroll
    for (int ky = 0; ky < 3; ++ky)
#pragma unroll
      for (int kx = 0; kx < 3; ++kx)
        acc += img[ic * 50176 + (y + ky) * 224 + (x + kx)] *
               w[((oc * 3 + ic) * 3 + ky) * 3 + kx];
  feats[idx] = fmaxf(acc, 0.0f);
}

// ------------------------------ PrimaryCaps conv 32->128 3x3 as implicit GEMM
// M = 16 output pixels (x-strip), N = 16 out channels, K = 288 (32ch x 3 x 3)
// WMMA f32 16x16x4: A 16x4 layout  K = 2*(lane>=16) + vgpr, M = lane&15
//                   B  4x16 layout K = 2*(lane>=16) + vgpr, N = lane&15
//                   D lanes0-15: M=vgpr, lanes16-31: M=vgpr+8; N = lane&15
__global__ void prim_conv_wmma(const float* __restrict__ feats,
                               const float* __restrict__ pw,
                               const float* __restrict__ pb,
                               float* __restrict__ u_pre) {
  __shared__ float patch[3 * 18 * 32];   // [dy][dxcol][c]
  const int tid = threadIdx.x;           // 128 threads = 4 waves
  const int x0  = blockIdx.x * 16;       // output x strip
  const int y   = blockIdx.y;            // output row

  // cooperative stage of the 18x3x32 input patch
  for (int t = tid; t < 3 * 18 * 32; t += 128) {
    int c    = t & 31;
    int rest = t >> 5;
    int dxc  = rest % 18;
    int dy   = rest / 18;
    int ix   = x0 + dxc;
    int iy   = y + dy;                   // <= 221, in range
    float v  = 0.0f;
    if (ix < 222) v = feats[(size_t)c * A1 + iy * 222 + ix];
    patch[t] = v;
  }
  __syncthreads();

  const int wave = tid >> 5;
  const int lane = tid & 31;
  const int m    = lane & 15;            // pixel index within strip / N index
  const int g    = lane >> 4;

  for (int nt = 0; nt < 2; ++nt) {
    const int ncbase = wave * 32 + nt * 16;
    const int oc     = ncbase + m;       // out-channel for B gather & store
    v8f acc = {};
    for (int kb = 0; kb < 288; kb += 4) {
      v2f av, bv;
#pragma unroll
      for (int r = 0; r < 2; ++r) {
        int k   = kb + g * 2 + r;
        int c   = k / 9;
        int rem = k - c * 9;
        int dy  = rem / 3;
        int dx  = rem - dy * 3;
        av[r] = patch[(dy * 18 + (m + dx)) * 32 + c];
        bv[r] = pw[((size_t)oc * 32 + c) * 9 + dy * 3 + dx];
      }
      acc = __builtin_amdgcn_wmma_f32_16x16x4_f32(
          false, av, false, bv, (short)0, acc, false, false);
    }
    const float bias = pb[oc];
#pragma unroll
    for (int vv = 0; vv < 8; ++vv) {
      int mm = vv + g * 8;
      int x  = x0 + mm;
      if (x < 220) u_pre[(size_t)oc * NP + y * 220 + x] = acc[vv] + bias;
    }
  }
}

// --------------------------- reshape (128,220,220)->(N,8) + squash over 8-dim
__global__ void squash_primary(const float* __restrict__ u_pre,
                               float* __restrict__ u) {
  int i = blockIdx.x * 256 + threadIdx.x;          // exact N (3025*256)
  int m = i / NP;
  int p = i - m * NP;
  float val[8];
  float sq = 0.0f;
#pragma unroll
  for (int k = 0; k < 8; ++k) {
    val[k] = u_pre[(size_t)(k * 16 + m) * NP + p];
    sq += val[k] * val[k];
  }
  float sc = sq / ((1.0f + sq) * sqrtf(sq));
#pragma unroll
  for (int k = 0; k < 8; ++k) u[(size_t)i * 8 + k] = val[k] * sc;
}

// ---------------- u_hat[j,i,d] = sum_k W[j,i,k,d]*u[i,k]; fused s1 = 0.1*sum_i
__global__ void uhat_kernel(const float* __restrict__ W,
                            const float* __restrict__ u,
                            float* __restrict__ uhat,
                            float* __restrict__ s1) {
  __shared__ float sPart[16];
  const int tid = threadIdx.x;
  if (tid < 16) sPart[tid] = 0.0f;
  __syncthreads();
  const int j  = blockIdx.y;
  const int i  = blockIdx.x * 64 + (tid >> 2);     // exact: 12100*64 = N
  const int dg = tid & 3;

  const float4* uv = (const float4*)(u + (size_t)i * 8);
  const float4 u0 = uv[0], u1 = uv[1];
  const float* wp = W + ((size_t)j * NNODES + i) * 128 + dg * 4;
  __builtin_prefetch(wp + 8192, 0, 1);             // stream-ahead hint

  float4 acc = {0.f, 0.f, 0.f, 0.f};
  float4 wv;
#define ACC(K, UK)                                                    \
  wv = *(const float4*)(wp + (K) * 16);                               \
  acc.x += (UK) * wv.x; acc.y += (UK) * wv.y;                         \
  acc.z += (UK) * wv.z; acc.w += (UK) * wv.w;
  ACC(0, u0.x) ACC(1, u0.y) ACC(2, u0.z) ACC(3, u0.w)
  ACC(4, u1.x) ACC(5, u1.y) ACC(6, u1.z) ACC(7, u1.w)
#undef ACC

  *(float4*)(uhat + ((size_t)j * NNODES + i) * 16 + dg * 4) = acc;

  atomAddL(&sPart[dg * 4 + 0], acc.x);
  atomAddL(&sPart[dg * 4 + 1], acc.y);
  atomAddL(&sPart[dg * 4 + 2], acc.z);
  atomAddL(&sPart[dg * 4 + 3], acc.w);
  __syncthreads();
  if (tid < 16) atomAddG(&s1[j * 16 + tid], sPart[tid] * 0.1f);
}

// ------------------------------------------------------- squash 10x16 vectors
__global__ void squash_vec(const float* __restrict__ s, float* __restrict__ v) {
  __shared__ float rowsq[10];
  int t = threadIdx.x;                             // 160
  if (t < 10) rowsq[t] = 0.0f;
  __syncthreads();
  float val = s[t];
  atomAddL(&rowsq[t >> 4], val * val);
  __syncthreads();
  float sq = rowsq[t >> 4];
  v[t] = val * (sq / ((1.0f + sq) * sqrtf(sq)));
}

// ----------- fused routing pass: b(+)=uhat.v ; c=softmax_j(b) ; s+=c*uhat
__global__ void routing_pass(const float* __restrict__ uhat,
                             float* __restrict__ blog,
                             const float* __restrict__ vprev,
                             float* __restrict__ sout,
                             int addB, int writeB) {
  __shared__ float vsh[160];
  __shared__ float sPart[160];
  int tid = threadIdx.x;                           // 256
  if (tid < 160) { vsh[tid] = vprev[tid]; sPart[tid] = 0.0f; }
  __syncthreads();
  const int i = blockIdx.x * 256 + tid;            // exact: 3025*256 = N
  const float* base = uhat + (size_t)i * 16;

  float t[10];
  for (int j = 0; j < 10; ++j) {
    const float4* p4 = (const float4*)(base + (size_t)j * JST);
    float4 a = p4[0], b4 = p4[1], c4 = p4[2], d4 = p4[3];
    const float* vv = vsh + j * 16;
    float dot = a.x * vv[0] + a.y * vv[1] + a.z * vv[2] + a.w * vv[3]
              + b4.x * vv[4] + b4.y * vv[5] + b4.z * vv[6] + b4.w * vv[7]
              + c4.x * vv[8] + c4.y * vv[9] + c4.z * vv[10] + c4.w * vv[11]
              + d4.x * vv[12] + d4.y * vv[13] + d4.z * vv[14] + d4.w * vv[15];
    if (addB) dot += blog[(size_t)j * NNODES + i];
    if (writeB) blog[(size_t)j * NNODES + i] = dot;
    t[j] = dot;
  }
  float mx = t[0];
  for (int j = 1; j < 10; ++j) mx = fmaxf(mx, t[j]);
  float Z = 0.0f;
  for (int j = 0; j < 10; ++j) { t[j] = expf(t[j] - mx); Z += t[j]; }
  float inv = 1.0f / Z;
  for (int j = 0; j < 10; ++j) {
    float c = t[j] * inv;
    const float4* p4 = (const float4*)(base + (size_t)j * JST);  // L2-hot reload
    float4 a = p4[0], b4 = p4[1], c4 = p4[2], d4 = p4[3];
    float* sp = sPart + j * 16;
    atomAddL(sp + 0, c * a.x);  atomAddL(sp + 1, c * a.y);
    atomAddL(sp + 2, c * a.z);  atomAddL(sp + 3, c * a.w);
    atomAddL(sp + 4, c * b4.x); atomAddL(sp + 5, c * b4.y);
    atomAddL(sp + 6, c * b4.z); atomAddL(sp + 7, c * b4.w);
    atomAddL(sp + 8, c * c4.x); atomAddL(sp + 9, c * c4.y);
    atomAddL(sp + 10, c * c4.z); atomAddL(sp + 11, c * c4.w);
    atomAddL(sp + 12, c * d4.x); atomAddL(sp + 13, c * d4.y);
    atomAddL(sp + 14, c * d4.z); atomAddL(sp + 15, c * d4.w);
  }
  __syncthreads();
  if (tid < 160) atomAddG(&sout[tid], sPart[tid]);
}

// ------------------- final squash, write v to d_out, pick winner, mask row
__global__ void finalize_caps(const float* __restrict__ s,
                              float* __restrict__ vout,
                              float* __restrict__ vmask,
                              int* __restrict__ winner) {
  __shared__ float rowsq[10];
  __shared__ int wsh;
  int t = threadIdx.x;                             // 160
  int j = t >> 4;
  if (t < 10) rowsq[t] = 0.0f;
  __syncthreads();
  float val = s[t];
  atomAddL(&rowsq[j], val * val);
  __syncthreads();
  float sq = rowsq[j];
  float vv = val * (sq / ((1.0f + sq) * sqrtf(sq)));
  vout[t] = vv;
  if (t == 0) {
    int w = 0;
    float best = rowsq[0];
    for (int jj = 1; jj < 10; ++jj)
      if (rowsq[jj] > best) { best = rowsq[jj]; w = jj; }
    wsh = w;
    *winner = w;
  }
  __syncthreads();
  if (j == wsh) vmask[t & 15] = vv;
}

// -------------------------------------------------------------- decoder MLPs
__global__ void dec1_kernel(const float* __restrict__ vmask,
                            const int* __restrict__ winner,
                            const float* __restrict__ w1,
                            const float* __restrict__ b1,
                            float* __restrict__ h1) {
  int o = blockIdx.x * 256 + threadIdx.x;          // 512
  int win = *winner;
  float acc = b1[o];
#pragma unroll
  for (int d = 0; d < 16; ++d)
    acc += vmask[d] * w1[(size_t)(win * 16 + d) * 512 + o];
  h1[o] = fmaxf(acc, 0.0f);
}

__global__ void dec2_kernel(const float* __restrict__ h1,
                            const float* __restrict__ w2,
                            const float* __restrict__ b2,
                            float* __restrict__ h2) {
  __shared__ float hs[512];
  int tid = threadIdx.x;
  for (int t = tid; t < 512; t += 256) hs[t] = h1[t];
  __syncthreads();
  int o = blockIdx.x * 256 + tid;                  // 1024
  float acc = b2[o];
  for (int h = 0; h < 512; ++h) acc += hs[h] * w2[(size_t)h * 1024 + o];
  h2[o] = fmaxf(acc, 0.0f);
}

__global__ void dec3_kernel(const float* __restrict__ h2,
                            const float* __restrict__ w3,
                            const float* __restrict__ b3,
                            float* __restrict__ recon) {
  __shared__ float hs[1024];
  int tid = threadIdx.x;
  for (int t = tid; t < 1024; t += 256) hs[t] = h2[t];
  __syncthreads();
  int o = blockIdx.x * 256 + tid;                  // 196*256 = 50176 exact
  float acc = b3[o];
  for (int h = 0; h < 1024; ++h) acc += hs[h] * w3[(size_t)h * 50176 + o];
  recon[o] = 1.0f / (1.0f + expf(-acc));
}

// ---------------------------------------------------------------------------
extern "C" void kernel_launch(void* const* d_in, const int* in_sizes, int n_in,
                              void* d_out, int out_size, void* d_ws, size_t ws_size,
                              hipStream_t stream) {
  const float* images = (const float*)d_in[0];
  const float* conv_w = (const float*)d_in[1];
  const float* conv_b = (const float*)d_in[2];
  const float* prim_w = (const float*)d_in[3];
  const float* prim_b = (const float*)d_in[4];
  const float* W      = (const float*)d_in[5];
  const float* d1w    = (const float*)d_in[6];
  const float* d1b    = (const float*)d_in[7];
  const float* d2w    = (const float*)d_in[8];
  const float* d2b    = (const float*)d_in[9];
  const float* d3w    = (const float*)d_in[10];
  const float* d3b    = (const float*)d_in[11];
  float* out = (float*)d_out;

  // workspace layout (floats); ~583 MB total, float4-aligned
  float* ws = (float*)d_ws;
  size_t off = 0;
  float* feats = ws + off; off += 32ull * A1;            // 1,577,088
  float* u_pre = ws + off; off += 128ull * NP;           // 6,195,200
  float* u     = ws + off; off += (size_t)NNODES * 8;    // 6,195,200
  float* blog  = ws + off; off += (size_t)10 * NNODES;   // 7,744,000
  float* s1    = ws + off; off += 160;
  float* s2    = ws + off; off += 160;
  float* s3    = ws + off; off += 160;
  float* v1    = ws + off; off += 160;
  float* v2    = ws + off; off += 160;
  float* vmask = ws + off; off += 16;
  int*   winner = (int*)(ws + off); off += 16;
  float* h1    = ws + off; off += 512;
  float* h2    = ws + off; off += 1024;
  float* uhat  = ws + off; off += (size_t)10 * NNODES * 16;  // 495.6 MB

  zero_kernel<<<1, 512, 0, stream>>>(s1, 480);           // s1,s2,s3 contiguous

  conv1_relu<<<(32 * A1 + 255) / 256, 256, 0, stream>>>(images, conv_w, conv_b, feats);

  dim3 pg(14, 220);                                      // 14 x-strips * 220 rows
  prim_conv_wmma<<<pg, 128, 0, stream>>>(feats, prim_w, prim_b, u_pre);

  squash_primary<<<NNODES / 256, 256, 0, stream>>>(u_pre, u);

  dim3 ug(NNODES / 64, 10);                              // 12100 x 10
  uhat_kernel<<<ug, 256, 0, stream>>>(W, u, uhat, s1);

  squash_vec<<<1, 160, 0, stream>>>(s1, v1);
  routing_pass<<<NNODES / 256, 256, 0, stream>>>(uhat, blog, v1, s2, 0, 1);
  squash_vec<<<1, 160, 0, stream>>>(s2, v2);
  routing_pass<<<NNODES / 256, 256, 0, stream>>>(uhat, blog, v2, s3, 1, 0);

  finalize_caps<<<1, 160, 0, stream>>>(s3, out, vmask, winner);

  dec1_kernel<<<2, 256, 0, stream>>>(vmask, winner, d1w, d1b, h1);
  dec2_kernel<<<4, 256, 0, stream>>>(h1, d2w, d2b, h2);
  dec3_kernel<<<196, 256, 0, stream>>>(h2, d3w, d3b, out + 160);
}